// LocalPatchMixer_52072183497306
// MI455X (gfx1250) — compile-verified
//
#include <hip/hip_runtime.h>
#include <cstdint>

// Problem constants (from reference): B=16, N=16384=128*128, D=256, K=3
#define B_   16
#define N_   16384
#define D_   256
#define HW_  128
#define PTS  8    // output points processed per wave (double-buffered TDM)

typedef unsigned int u32x4 __attribute__((ext_vector_type(4)));
typedef int          i32x8 __attribute__((ext_vector_type(8)));
typedef int          i32x4 __attribute__((ext_vector_type(4)));

__device__ __forceinline__ float gelu_exact(float v) {
  // exact (erf) GELU, matching jax.nn.gelu(approximate=False)
  return 0.5f * v * (1.0f + erff(v * 0.70710678118654752440f));
}

// ---------------------------------------------------------------------------
// Issue one TDM gather: 9 neighbor rows (1 KiB each) of x[b] -> LDS buffer.
// Gather-mode descriptor, 16-bit row indices taken from inv[] (scalar loads,
// all blockIdx-uniform since blockDim == 32). Returns 9-bit validity mask
// (border taps use row 0 and are skipped in the accumulation).
// ---------------------------------------------------------------------------
__device__ __forceinline__ unsigned issue_gather(
    const float* __restrict__ xb_base, const int* __restrict__ coords,
    const int* __restrict__ inv, int b, int p, uint32_t lds_off) {
  const int gx = __builtin_amdgcn_readfirstlane(coords[2 * p + 0]);
  const int gy = __builtin_amdgcn_readfirstlane(coords[2 * p + 1]);

  unsigned vm = 0;
  unsigned idx[9];
#pragma unroll
  for (int t = 0; t < 9; ++t) {
    const int  dy = t / 3 - 1, dx = t % 3 - 1;
    const int  yy = gy + dy, xx = gx + dx;
    const bool v  = ((unsigned)yy < HW_) && ((unsigned)xx < HW_);
    if (v) vm |= (1u << t);
    const int r = v ? inv[(b << 14) + (yy << 7) + xx] : 0;
    idx[t]      = (unsigned)__builtin_amdgcn_readfirstlane(r);
  }

  const uint64_t gaddr = (uint64_t)(uintptr_t)xb_base;

  u32x4 g0;
  g0[0] = 0x80000001u;  // count=1, is_restore=0, gather_mode=1, 16-bit indices
  g0[1] = lds_off;                                // lds_addr (bytes)
  g0[2] = (uint32_t)gaddr;                        // global_addr[31:0]
  g0[3] = ((uint32_t)(gaddr >> 32) & 0x01FFFFFFu) // global_addr[56:32]
          | (2u << 30);                           // type = 2 ("image")

  i32x8 g1;
  g1[0] = (int)(2u << 16);      // workgroup_mask=0, data_size=2 (4 bytes)
  g1[1] = (int)(256u << 16);    // tensor_dim0[15:0] = 256 (row length, elems)
  g1[2] = (int)(16384u << 16);  // tensor_dim0 hi=0 | tensor_dim1[15:0]=16384
  g1[3] = (int)(256u << 16);    // tensor_dim1 hi=0 | tile_dim0 = 256
  g1[4] = 9;                    // tile_dim1 = #valid gather indices = 9
  g1[5] = 256;                  // tensor_dim0_stride = 256 elements
  g1[6] = 0;
  g1[7] = 0;

  i32x4 g2;  // gather row indices 0..7 (16-bit pairs)
  g2[0] = (int)(idx[0] | (idx[1] << 16));
  g2[1] = (int)(idx[2] | (idx[3] << 16));
  g2[2] = (int)(idx[4] | (idx[5] << 16));
  g2[3] = (int)(idx[6] | (idx[7] << 16));
  i32x4 g3;  // gather row index 8 (+ ignored slots)
  g3[0] = (int)(idx[8] | (idx[8] << 16));
  g3[1] = 0; g3[2] = 0; g3[3] = 0;

  // WAR: the target LDS buffer was read by ds_load_b128 two iterations ago;
  // the TDM writes LDS outside the compiler's dependence model.
  asm volatile("s_wait_dscnt 0x0" ::: "memory");

#if __clang_major__ >= 23
  {  // amdgpu-toolchain (therock-10.0 headers): 6-arg form
    i32x8 gpad = {};
    __builtin_amdgcn_tensor_load_to_lds(g0, g1, g2, g3, gpad, 0);
  }
#else
  // ROCm 7.2 / clang-22: 5-arg form
  __builtin_amdgcn_tensor_load_to_lds(g0, g1, g2, g3, 0);
#endif
  return vm;
}

// ---------------------------------------------------------------------------
// Depthwise 3x3 accumulate from one gathered 9-row LDS buffer, + bias,
// exact GELU, residual add, coalesced b128 store. Weights preloaded in VGPRs.
// ---------------------------------------------------------------------------
__device__ __forceinline__ void compute_point(
    const float* __restrict__ x, float* __restrict__ out,
    const float* __restrict__ srow, const float (&wr)[8][9],
    const float (&br)[8], unsigned vm, int p, int lane) {
  float acc[8];
#pragma unroll
  for (int i = 0; i < 8; ++i) acc[i] = br[i];

  const float4* s4 = reinterpret_cast<const float4*>(srow);
#pragma unroll
  for (int t = 0; t < 9; ++t) {
    if (vm & (1u << t)) {  // wave-uniform branch
      const float4 va = s4[t * 64 + lane];       // channels lane*4 .. +3
      const float4 vb = s4[t * 64 + 32 + lane];  // channels 128+lane*4 .. +3
      acc[0] += wr[0][t] * va.x;
      acc[1] += wr[1][t] * va.y;
      acc[2] += wr[2][t] * va.z;
      acc[3] += wr[3][t] * va.w;
      acc[4] += wr[4][t] * vb.x;
      acc[5] += wr[5][t] * vb.y;
      acc[6] += wr[6][t] * vb.z;
      acc[7] += wr[7][t] * vb.w;
    }
  }

  const float4* xr = reinterpret_cast<const float4*>(x + (size_t)p * D_);
  const float4  xa = xr[lane];
  const float4  xb = xr[32 + lane];
  float4 oa, ob;
  oa.x = xa.x + gelu_exact(acc[0]);
  oa.y = xa.y + gelu_exact(acc[1]);
  oa.z = xa.z + gelu_exact(acc[2]);
  oa.w = xa.w + gelu_exact(acc[3]);
  ob.x = xb.x + gelu_exact(acc[4]);
  ob.y = xb.y + gelu_exact(acc[5]);
  ob.z = xb.z + gelu_exact(acc[6]);
  ob.w = xb.w + gelu_exact(acc[7]);

  float4* o4 = reinterpret_cast<float4*>(out + (size_t)p * D_);
  o4[lane]      = oa;
  o4[32 + lane] = ob;
}

// ---------------------------------------------------------------------------
// Main kernel: one wave (32 lanes) per PTS consecutive points of one batch.
// Per wave: preload this lane's 72 weights + 8 biases once, then a software-
// pipelined loop: TDM-gather point i+1 into the other LDS buffer while
// computing point i (s_wait_tensorcnt 1 = previous gather complete, since
// TDM ops from one wave complete in order).
// ---------------------------------------------------------------------------
__global__ __launch_bounds__(32) void mixer_kernel(
    const float* __restrict__ x, const int* __restrict__ coords,
    const float* __restrict__ conv_w, const float* __restrict__ conv_b,
    const int* __restrict__ inv, float* __restrict__ out) {
  __shared__ float smem[2][9 * D_];  // double-buffered 9-row gather targets

  const int p0   = blockIdx.x * PTS;  // uniform; PTS divides N -> same batch
  const int b    = p0 >> 14;
  const int lane = (int)threadIdx.x;

  const float* xb_base = x + (size_t)b * N_ * D_;

  // Preload this lane's 8 channels of weights/bias (amortized over PTS points)
  const int ca = lane * 4;
  float wr[8][9];
  float br[8];
#pragma unroll
  for (int i = 0; i < 4; ++i) {
    br[i]     = conv_b[ca + i];
    br[4 + i] = conv_b[128 + ca + i];
#pragma unroll
    for (int t = 0; t < 9; ++t) {
      wr[i][t]     = conv_w[(ca + i) * 9 + t];
      wr[4 + i][t] = conv_w[(128 + ca + i) * 9 + t];
    }
  }

  const uint32_t lds0 = (uint32_t)(uintptr_t)(&smem[0][0]);
  const uint32_t bufb = (uint32_t)(9 * D_ * sizeof(float));  // 9 KiB

  unsigned vm[PTS];
  vm[0] = issue_gather(xb_base, coords, inv, b, p0, lds0);

#pragma unroll
  for (int i = 0; i < PTS; ++i) {
    if (i + 1 < PTS) {
      vm[i + 1] = issue_gather(xb_base, coords, inv, b, p0 + i + 1,
                               lds0 + ((unsigned)(i + 1) & 1u) * bufb);
      __builtin_amdgcn_s_wait_tensorcnt(1);  // buffer i ready; i+1 in flight
    } else {
      __builtin_amdgcn_s_wait_tensorcnt(0);  // last buffer ready
    }
    asm volatile("" ::: "memory");  // TDM wrote LDS behind the compiler's back
    compute_point(x, out, &smem[i & 1][0], wr, br, vm[i], p0 + i, lane);
  }
}

// ---------------------------------------------------------------------------
// Kernel 1: build inverse permutation. coords is a per-batch permutation of
// the 128x128 grid (collision-free scatter), so every slot is written exactly
// once and per-batch min is (0,0) -> normalization is identity.
// inv[b, gy*128+gx] = n
// ---------------------------------------------------------------------------
__global__ void build_inv_kernel(const int* __restrict__ coords,
                                 int* __restrict__ inv) {
  const int i  = blockIdx.x * blockDim.x + threadIdx.x;  // 0 .. B*N-1
  const int b  = i >> 14;
  const int n  = i & (N_ - 1);
  const int gx = coords[2 * i + 0];
  const int gy = coords[2 * i + 1];
  inv[(b << 14) + (gy << 7) + gx] = n;
}

// ---------------------------------------------------------------------------
extern "C" void kernel_launch(void* const* d_in, const int* in_sizes, int n_in,
                              void* d_out, int out_size, void* d_ws, size_t ws_size,
                              hipStream_t stream) {
  (void)in_sizes; (void)n_in; (void)out_size; (void)ws_size;
  const float* x      = (const float*)d_in[0];  // (B, N, D) f32
  const int*   coords = (const int*)d_in[1];    // (B, N, 2) i32
  const float* conv_w = (const float*)d_in[2];  // (D, 1, 3, 3) f32
  const float* conv_b = (const float*)d_in[3];  // (D,) f32
  float*       out    = (float*)d_out;          // (B, N, D) f32
  int*         inv    = (int*)d_ws;             // B*N i32 = 1 MiB scratch

  // Phase 1: inverse permutation (fully overwritten -> no init needed)
  build_inv_kernel<<<(B_ * N_) / 256, 256, 0, stream>>>(coords, inv);

  // Phase 2: PTS points per wave; batch-ordered blocks keep the 16 MiB
  // per-batch x slice resident in the 192 MB L2 across the 9x TDM reuse.
  mixer_kernel<<<(B_ * N_) / PTS, 32, 0, stream>>>(x, coords, conv_w, conv_b,
                                                   inv, out);
}